// HTGNN_29231547417245
// MI455X (gfx1250) — compile-verified
//
#include <hip/hip_runtime.h>

static constexpr int   NN   = 100000;          // nodes
static constexpr int   DC   = 128;             // channels
static constexpr int   EE   = 320000;          // edges per relation
static constexpr int   ELE  = 200000;          // label edges per relation
static constexpr int   TOTE = EE + NN;         // edges incl. self loops
static constexpr size_t NF  = (size_t)NN * DC;

typedef __attribute__((ext_vector_type(2))) float v2f;
typedef __attribute__((ext_vector_type(8))) float v8f;
typedef unsigned int u32x4 __attribute__((ext_vector_type(4)));
typedef int          i32x4 __attribute__((ext_vector_type(4)));
typedef int          i32x8 __attribute__((ext_vector_type(8)));

// ---------------- ordered-float helpers for atomic max --------------------
__device__ __forceinline__ unsigned f2ord(float f) {
  unsigned b = __float_as_uint(f);
  return (b & 0x80000000u) ? ~b : (b | 0x80000000u);
}
__device__ __forceinline__ float ord2f(unsigned u) {
  return __uint_as_float((u & 0x80000000u) ? (u ^ 0x80000000u) : ~u);
}

// ---------------- TDM: 2D tile global -> LDS with row padding -------------
// Loads tile (tile_d0 x tile_d1) of a (tensor_d0 x tensor_d1) f32 tensor,
// row stride stride_d0 elements, into LDS at ldsOff with +4 DWORD padding
// after every 128 DWORDs (i.e. LDS row stride 132 floats). OOB rows -> 0.
__device__ __forceinline__ void tdm_load_2d(unsigned ldsOff, const void* gaddr,
                                            unsigned tensor_d0, unsigned tensor_d1,
                                            unsigned tile_d0, unsigned tile_d1,
                                            unsigned stride_d0)
{
  const unsigned long long ga = (unsigned long long)(size_t)gaddr;
  u32x4 g0;
  g0.x = 1u;                                           // count=1, user descriptor
  g0.y = ldsOff;                                       // lds_addr (bytes)
  g0.z = (unsigned)(ga & 0xFFFFFFFFu);                 // global_addr[31:0]
  g0.w = (unsigned)((ga >> 32) & 0x01FFFFFFu) | 0x80000000u;  // addr[56:32] | type=2
  i32x8 g1;
  g1[0] = (int)((2u << 16)        // data_size = 4B
              | (1u << 20)        // pad_enable
              | (6u << 22)        // pad_interval: 128 DWORDs
              | (3u << 25));      // pad_amount: 4 DWORDs
  g1[1] = (int)((tensor_d0 & 0xFFFFu) << 16);
  g1[2] = (int)(((tensor_d0 >> 16) & 0xFFFFu) | ((tensor_d1 & 0xFFFFu) << 16));
  g1[3] = (int)(((tensor_d1 >> 16) & 0xFFFFu) | ((tile_d0 & 0xFFFFu) << 16));
  g1[4] = (int)(tile_d1 & 0xFFFFu);                    // tile_dim1 (tile_dim2 = 0)
  g1[5] = (int)stride_d0;                              // tensor_dim0_stride[31:0]
  g1[6] = 0;                                           // stride hi / dim1_stride lo
  g1[7] = 0;
  const i32x4 z = {0, 0, 0, 0};
#if defined(__clang_major__) && (__clang_major__ >= 23)
  const i32x8 z8 = {0, 0, 0, 0, 0, 0, 0, 0};
  __builtin_amdgcn_tensor_load_to_lds(g0, g1, z, z, z8, 0);
#else
  __builtin_amdgcn_tensor_load_to_lds(g0, g1, z, z, 0);
#endif
}

// ---------------- GEMM: C[M,128] = A[M,128] @ W[128,128] (+bias) ----------
// Block: 256 threads (8 waves) -> 128 rows x 128 cols.
// Wave w: rows (w>>1)*32..+31 (2 row tiles), col half (w&1) (4 col tiles).
// 8x v8f accumulators; V_WMMA_F32_16X16X4_F32, K fully unrolled.
// Tiles staged by the Tensor Data Mover (wave 0), LDS stride 132 (padded).
__global__ __launch_bounds__(256) void gemm128_wmma(
    const float* __restrict__ A, const float* __restrict__ W,
    const float* __restrict__ bias, float* __restrict__ C, int M)
{
  __shared__ float Wl[128 * 132];   // 67.6 KB
  __shared__ float Al[128 * 132];   // 67.6 KB
  const int tid  = threadIdx.x;
  const int lane = tid & 31;
  const int wave = tid >> 5;
  const int rowBase = blockIdx.x * 128;

  if (wave == 0) {
    // W panel: 128x128, dense
    tdm_load_2d((unsigned)(size_t)&Wl[0], W, 128u, 128u, 128u, 128u, 128u);
    // A tile: up to 128 rows; rows past M read as zero via tensor OOB
    tdm_load_2d((unsigned)(size_t)&Al[0], A + (size_t)rowBase * 128,
                128u, (unsigned)(M - rowBase), 128u, 128u, 128u);
    __builtin_amdgcn_s_wait_tensorcnt(0);
  }
  __syncthreads();

  const int rowHalf = (wave >> 1) * 32;   // 0,32,64,96
  const int colBase = (wave & 1) * 64;    // 0 / 64
  const int kh      = (lane >> 4) << 1;   // 0 or 2
  const int nlane   = lane & 15;
  const int mrow0   = (rowHalf + (lane & 15)) * 132;
  const int mrow1   = mrow0 + 16 * 132;

  v8f acc[8] = {};
#pragma unroll
  for (int kk = 0; kk < 128; kk += 4) {
    v2f a0, a1;
    a0.x = Al[mrow0 + kk + kh];
    a0.y = Al[mrow0 + kk + kh + 1];
    a1.x = Al[mrow1 + kk + kh];
    a1.y = Al[mrow1 + kk + kh + 1];
    const float* wr0 = &Wl[(kk + kh) * 132 + colBase + nlane];
    const float* wr1 = wr0 + 132;
#pragma unroll
    for (int j = 0; j < 4; ++j) {
      v2f b;
      b.x = wr0[j * 16];
      b.y = wr1[j * 16];
      acc[j]     = __builtin_amdgcn_wmma_f32_16x16x4_f32(
                       false, a0, false, b, (short)0, acc[j],     false, false);
      acc[4 + j] = __builtin_amdgcn_wmma_f32_16x16x4_f32(
                       false, a1, false, b, (short)0, acc[4 + j], false, false);
    }
  }

  // C/D layout: vgpr r -> lanes 0-15: M=r, lanes 16-31: M=r+8; N = lane&15
  const int mOut0 = rowBase + rowHalf + ((lane >> 4) << 3);
#pragma unroll
  for (int j = 0; j < 4; ++j) {
    const int col = colBase + j * 16 + nlane;
    const float badd = bias ? bias[col] : 0.f;
#pragma unroll
    for (int r = 0; r < 8; ++r) {
      const int row0 = mOut0 + r;
      const int row1 = row0 + 16;
      if (row0 < M) C[(size_t)row0 * 128 + col] = acc[j][r] + badd;
      if (row1 < M) C[(size_t)row1 * 128 + col] = acc[4 + j][r] + badd;
    }
  }
}

// ---------------- GAT kernels --------------------------------------------
__global__ void gat_init(float* __restrict__ out, const float* __restrict__ bias,
                         unsigned* __restrict__ mx, float* __restrict__ den)
{
  const int n = blockIdx.x, c = threadIdx.x;
  out[(size_t)n * 128 + c] = bias[c];
  if (c == 0) { mx[n] = 0u; den[n] = 0.f; }
}

// wave per edge: e = att . leaky_relu(hl[src] + hr[dst], 0.2); atomic seg-max
__global__ __launch_bounds__(256) void gat_scores(
    const int* __restrict__ ei, const float* __restrict__ hl,
    const float* __restrict__ hr, const float* __restrict__ att,
    float* __restrict__ ev, unsigned* __restrict__ mx)
{
  const int e    = (blockIdx.x * blockDim.x + threadIdx.x) >> 5;
  const int lane = threadIdx.x & 31;
  if (e >= TOTE) return;
  int s, d;
  if (e < EE) { s = ei[e]; d = ei[EE + e]; } else { s = d = e - EE; }
  const float4 l4 = *(const float4*)(hl + (size_t)s * 128 + lane * 4);
  const float4 r4 = *(const float4*)(hr + (size_t)d * 128 + lane * 4);
  const float4 a4 = *(const float4*)(att + lane * 4);
  float v = 0.f, t;
  t = l4.x + r4.x; v += a4.x * (t > 0.f ? t : 0.2f * t);
  t = l4.y + r4.y; v += a4.y * (t > 0.f ? t : 0.2f * t);
  t = l4.z + r4.z; v += a4.z * (t > 0.f ? t : 0.2f * t);
  t = l4.w + r4.w; v += a4.w * (t > 0.f ? t : 0.2f * t);
#pragma unroll
  for (int off = 16; off; off >>= 1) v += __shfl_xor(v, off, 32);
  if (lane == 0) { ev[e] = v; atomicMax(&mx[d], f2ord(v)); }
}

__global__ void gat_decode_max(unsigned* __restrict__ mx, int n)
{
  const int i = blockIdx.x * blockDim.x + threadIdx.x;
  if (i < n) ((float*)mx)[i] = ord2f(mx[i]);
}

__global__ void gat_expden(const int* __restrict__ ei, float* __restrict__ ev,
                           const float* __restrict__ fmax, float* __restrict__ den)
{
  const int e = blockIdx.x * blockDim.x + threadIdx.x;
  if (e >= TOTE) return;
  const int d = (e < EE) ? ei[EE + e] : e - EE;
  const float ex = __expf(ev[e] - fmax[d]);
  ev[e] = ex;
  atomicAdd(&den[d], ex);
}

// wave per edge: out[dst] += (ev/den[dst]) * hl[src]
__global__ __launch_bounds__(256) void gat_agg(
    const int* __restrict__ ei, const float* __restrict__ ev,
    const float* __restrict__ den, const float* __restrict__ hl,
    float* __restrict__ out)
{
  const int e    = (blockIdx.x * blockDim.x + threadIdx.x) >> 5;
  const int lane = threadIdx.x & 31;
  if (e >= TOTE) return;
  int s, d;
  if (e < EE) { s = ei[e]; d = ei[EE + e]; } else { s = d = e - EE; }
  const float alpha = ev[e] / den[d];
  const float4 l4 = *(const float4*)(hl + (size_t)s * 128 + lane * 4);
  float* o = out + (size_t)d * 128 + lane * 4;
  atomicAdd(o + 0, alpha * l4.x);
  atomicAdd(o + 1, alpha * l4.y);
  atomicAdd(o + 2, alpha * l4.z);
  atomicAdd(o + 3, alpha * l4.w);
}

__global__ void relu_inplace(float* __restrict__ x, size_t n)
{
  const size_t i = (size_t)blockIdx.x * blockDim.x + threadIdx.x;
  if (i < n) x[i] = fmaxf(x[i], 0.f);
}

// ---------------- semantic attention -------------------------------------
__global__ void zero4(float* __restrict__ s) { if (threadIdx.x < 4) s[threadIdx.x] = 0.f; }

__global__ __launch_bounds__(256) void sem_score(
    const float* __restrict__ t, const float* __restrict__ q,
    float* __restrict__ score)
{
  const int n    = (blockIdx.x * blockDim.x + threadIdx.x) >> 5;
  const int lane = threadIdx.x & 31;
  if (n >= NN) return;
  const float4 t4 = *(const float4*)(t + (size_t)n * 128 + lane * 4);
  const float4 q4 = *(const float4*)(q + lane * 4);
  float v = q4.x * tanhf(t4.x) + q4.y * tanhf(t4.y) +
            q4.z * tanhf(t4.z) + q4.w * tanhf(t4.w);
#pragma unroll
  for (int off = 16; off; off >>= 1) v += __shfl_xor(v, off, 32);
  if (lane == 0) atomicAdd(score, v * (1.f / (float)NN));
}

__global__ void softmax2(const float* __restrict__ s, float* __restrict__ a)
{
  const float m  = fmaxf(s[0], s[1]);
  const float e0 = __expf(s[0] - m), e1 = __expf(s[1] - m);
  const float inv = 1.f / (e0 + e1);
  a[0] = e0 * inv; a[1] = e1 * inv;
}

// agg = a0*m0 + a1*m1; cp = agg + enc(t=1); pp = past + enc(t=0)
__global__ void combine_pe(const float* __restrict__ m0, const float* __restrict__ m1,
                           const float* __restrict__ a, const float* __restrict__ past,
                           float* __restrict__ cp, float* __restrict__ pp)
{
  const size_t i = (size_t)blockIdx.x * blockDim.x + threadIdx.x;
  if (i >= NF) return;
  const int c = (int)(i & 127);
  const float agg = a[0] * m0[i] + a[1] * m1[i];
  const float invden = __powf(1000.f, -(float)c / 64.f);   // 1/denom
  const float enc1 = ((c & 1) == 0) ? __sinf(invden) : __cosf(invden);
  const float enc0 = ((c & 1) == 0) ? 0.f : 1.f;           // sin(0)/cos(0)
  cp[i] = agg + enc1;
  pp[i] = past[i] + enc0;
}

// cur = delta*leaky_relu(q*k*v,0.01) + (1-delta)*xlin
__global__ void final_combine(const float* __restrict__ q, const float* __restrict__ k,
                              const float* __restrict__ v, const float* __restrict__ xlin,
                              const float* __restrict__ delta, float* __restrict__ cur)
{
  const size_t i = (size_t)blockIdx.x * blockDim.x + threadIdx.x;
  if (i >= NF) return;
  const float p = q[i] * k[i] * v[i];
  const float u = (p > 0.f) ? p : 0.01f * p;
  const float dd = delta[0];
  cur[i] = dd * u + (1.f - dd) * xlin[i];
}

// ---------------- link logits --------------------------------------------
__global__ __launch_bounds__(256) void link_logits(
    const int* __restrict__ el, const float* __restrict__ cur2,
    const float* __restrict__ pW, const float* __restrict__ pb,
    float* __restrict__ out)
{
  const int e    = (blockIdx.x * blockDim.x + threadIdx.x) >> 5;
  const int lane = threadIdx.x & 31;
  if (e >= ELE) return;
  const int s = el[e], d = el[ELE + e];
  const int c = lane * 4;
  const float4 cs = *(const float4*)(cur2 + (size_t)s * 128 + c);
  const float4 cd = *(const float4*)(cur2 + (size_t)d * 128 + c);
  float v = 0.f;
#pragma unroll
  for (int j = 0; j < 4; ++j) {
    const float wsrc = pW[2 * (c + j)] + pW[2 * (c + j) + 1];
    const float wdst = pW[2 * (128 + c + j)] + pW[2 * (128 + c + j) + 1];
    const float es = (j == 0 ? cs.x : j == 1 ? cs.y : j == 2 ? cs.z : cs.w);
    const float ed = (j == 0 ? cd.x : j == 1 ? cd.y : j == 2 ? cd.z : cd.w);
    v += es * wsrc + ed * wdst;
  }
#pragma unroll
  for (int off = 16; off; off >>= 1) v += __shfl_xor(v, off, 32);
  if (lane == 0) out[e] = v + pb[0] + pb[1];
}

// ---------------- host side ----------------------------------------------
struct LayerP {
  const float *aggkW, *aggkb, *aggq, *delta;
  const float *g0Wl, *g0Wr, *g0att, *g0b;
  const float *g1Wl, *g1Wr, *g1att, *g1b;
  const float *linW, *linb;
  const float *updkW, *updkb, *updqW, *updqb, *updvW, *updvb;
};

static LayerP layer_params(void* const* d_in, int b) {
  LayerP p;
  p.aggkW = (const float*)d_in[b + 0];  p.aggkb = (const float*)d_in[b + 1];
  p.aggq  = (const float*)d_in[b + 2];  p.delta = (const float*)d_in[b + 3];
  p.g0Wl  = (const float*)d_in[b + 4];  p.g0Wr  = (const float*)d_in[b + 5];
  p.g0att = (const float*)d_in[b + 6];  p.g0b   = (const float*)d_in[b + 7];
  p.g1Wl  = (const float*)d_in[b + 8];  p.g1Wr  = (const float*)d_in[b + 9];
  p.g1att = (const float*)d_in[b + 10]; p.g1b   = (const float*)d_in[b + 11];
  p.linW  = (const float*)d_in[b + 12]; p.linb  = (const float*)d_in[b + 13];
  p.updkW = (const float*)d_in[b + 14]; p.updkb = (const float*)d_in[b + 15];
  p.updqW = (const float*)d_in[b + 16]; p.updqb = (const float*)d_in[b + 17];
  p.updvW = (const float*)d_in[b + 18]; p.updvb = (const float*)d_in[b + 19];
  return p;
}

static void run_gat(const int* ei, const float* hl, const float* hr,
                    const float* att, const float* bias, float* out,
                    float* EV, float* DEN, unsigned* MX, hipStream_t st)
{
  const int WB = (TOTE + 7) / 8;            // wave-per-edge, 8 waves/block
  gat_init<<<NN, 128, 0, st>>>(out, bias, MX, DEN);
  gat_scores<<<WB, 256, 0, st>>>(ei, hl, hr, att, EV, MX);
  gat_decode_max<<<(NN + 255) / 256, 256, 0, st>>>(MX, NN);
  gat_expden<<<(TOTE + 255) / 256, 256, 0, st>>>(ei, EV, (const float*)MX, DEN);
  gat_agg<<<WB, 256, 0, st>>>(ei, EV, DEN, hl, out);
  relu_inplace<<<(int)((NF + 255) / 256), 256, 0, st>>>(out, NF);
}

static void run_layer(const float* x, const float* past, const LayerP& P,
                      const int* ei0, const int* ei1, float* cur,
                      float* W0, float* W1, float* W2, float* W3,
                      float* W4, float* W5, float* W6,
                      float* EV, float* DEN, unsigned* MX, float* SC,
                      hipStream_t st)
{
  const int GB  = (NN + 127) / 128;
  const int EWB = (int)((NF + 255) / 256);
  // 5 input GEMMs
  gemm128_wmma<<<GB, 256, 0, st>>>(x, P.g0Wl, nullptr, W0, NN);  // hl0
  gemm128_wmma<<<GB, 256, 0, st>>>(x, P.g0Wr, nullptr, W1, NN);  // hr0
  gemm128_wmma<<<GB, 256, 0, st>>>(x, P.g1Wl, nullptr, W2, NN);  // hl1
  gemm128_wmma<<<GB, 256, 0, st>>>(x, P.g1Wr, nullptr, W3, NN);  // hr1
  gemm128_wmma<<<GB, 256, 0, st>>>(x, P.linW, P.linb, W4, NN);   // xlin
  // GATv2 per relation
  run_gat(ei0, W0, W1, P.g0att, P.g0b, W5, EV, DEN, MX, st);     // m0
  run_gat(ei1, W2, W3, P.g1att, P.g1b, W6, EV, DEN, MX, st);     // m1
  // semantic attention
  zero4<<<1, 32, 0, st>>>(SC);
  gemm128_wmma<<<GB, 256, 0, st>>>(W5, P.aggkW, P.aggkb, W0, NN);
  sem_score<<<(NN + 7) / 8, 256, 0, st>>>(W0, P.aggq, &SC[0]);
  gemm128_wmma<<<GB, 256, 0, st>>>(W6, P.aggkW, P.aggkb, W0, NN);
  sem_score<<<(NN + 7) / 8, 256, 0, st>>>(W0, P.aggq, &SC[1]);
  softmax2<<<1, 1, 0, st>>>(SC, SC + 2);
  // positional encodings: cp -> W1, pp -> W2
  combine_pe<<<EWB, 256, 0, st>>>(W5, W6, SC + 2, past, W1, W2);
  // update: q -> W5, k -> W6, v -> W3
  gemm128_wmma<<<GB, 256, 0, st>>>(W1, P.updqW, P.updqb, W5, NN);
  gemm128_wmma<<<GB, 256, 0, st>>>(W2, P.updkW, P.updkb, W6, NN);
  gemm128_wmma<<<GB, 256, 0, st>>>(W1, P.updvW, P.updvb, W3, NN);
  final_combine<<<EWB, 256, 0, st>>>(W5, W6, W3, W4, P.delta, cur);
}

extern "C" void kernel_launch(void* const* d_in, const int* in_sizes, int n_in,
                              void* d_out, int out_size, void* d_ws, size_t ws_size,
                              hipStream_t stream)
{
  (void)in_sizes; (void)n_in; (void)out_size; (void)ws_size;
  const float* x     = (const float*)d_in[0];
  const int*   ei0   = (const int*)d_in[1];
  const int*   ei1   = (const int*)d_in[2];
  const int*   el0   = (const int*)d_in[3];
  const int*   el1   = (const int*)d_in[4];
  const float* past1 = (const float*)d_in[5];
  const float* past2 = (const float*)d_in[6];
  const LayerP L1 = layer_params(d_in, 7);
  const LayerP L2 = layer_params(d_in, 27);
  const float* postW = (const float*)d_in[47];
  const float* postb = (const float*)d_in[48];

  float* out = (float*)d_out;
  float* h0   = out;
  float* h1   = out + ELE;
  float* cur1 = out + 2 * (size_t)ELE;
  float* cur2 = cur1 + NF;

  // workspace carve (7 node-feature buffers + edge/node scratch)
  float* ws = (float*)d_ws;
  float* W0 = ws;            float* W1 = W0 + NF;  float* W2 = W1 + NF;
  float* W3 = W2 + NF;       float* W4 = W3 + NF;  float* W5 = W4 + NF;
  float* W6 = W5 + NF;
  float*    EV  = W6 + NF;                 // TOTE floats
  float*    DEN = EV + TOTE;               // NN floats
  unsigned* MX  = (unsigned*)(DEN + NN);   // NN uints
  float*    SC  = (float*)(MX + NN);       // s0,s1,a0,a1

  run_layer(x,    past1, L1, ei0, ei1, cur1, W0, W1, W2, W3, W4, W5, W6,
            EV, DEN, MX, SC, stream);
  run_layer(cur1, past2, L2, ei0, ei1, cur2, W0, W1, W2, W3, W4, W5, W6,
            EV, DEN, MX, SC, stream);

  const int LB = (ELE + 7) / 8;
  link_logits<<<LB, 256, 0, stream>>>(el0, cur2, postW, postb, h0);
  link_logits<<<LB, 256, 0, stream>>>(el1, cur2, postW, postb, h1);
}